// ImprovedFaceMetricLoss_62947040690558
// MI455X (gfx1250) — compile-verified
//
#include <hip/hip_runtime.h>
#include <hip/hip_bf16.h>
#include <math.h>

typedef float v2f __attribute__((ext_vector_type(2)));
typedef float v8f __attribute__((ext_vector_type(8)));

#define B_ROWS 4096
#define DIM    512
#define SA_STRIDE 516   // pad: 516 % 64 = 4 -> conflict-free row access; 516 % 4 == 0 -> float4-aligned

// ---------------------------------------------------------------------------
// ws[0] = triplet sum, ws[1] = local sum, ws[2..2+4095] = per-row sum of squares
// ---------------------------------------------------------------------------

__global__ void zero_kernel(float* ws) {
    ws[0] = 0.0f;
    ws[1] = 0.0f;
}

// Per-row sum of squares: one wave32 per row, 8 rows per 256-thread block.
__global__ __launch_bounds__(256) void sq_kernel(const float* __restrict__ x,
                                                 float* __restrict__ sq_out) {
    const int lane = threadIdx.x & 31;
    const int wave = threadIdx.x >> 5;
    const int row  = blockIdx.x * 8 + wave;
    const float* xr = x + (size_t)row * DIM;
    float s = 0.0f;
    #pragma unroll
    for (int c = lane * 4; c < DIM; c += 128) {
        float4 v = *(const float4*)(xr + c);
        s += v.x * v.x + v.y * v.y + v.z * v.z + v.w * v.w;
    }
    #pragma unroll
    for (int off = 16; off > 0; off >>= 1) s += __shfl_xor(s, off, 32);
    if (lane == 0) sq_out[row] = s;
}

// Triplet hinge: norm(a-p+eps) - norm(a-n+eps) + 1, clamped, summed.
__global__ __launch_bounds__(256) void triplet_kernel(const float* __restrict__ a,
                                                      const float* __restrict__ p,
                                                      const float* __restrict__ n,
                                                      float* __restrict__ tsum) {
    const int lane = threadIdx.x & 31;
    const int wave = threadIdx.x >> 5;
    const int row  = blockIdx.x * 8 + wave;
    const size_t base = (size_t)row * DIM;
    float sp = 0.0f, sn = 0.0f;
    #pragma unroll
    for (int c = lane * 4; c < DIM; c += 128) {
        float4 av = *(const float4*)(a + base + c);
        float4 pv = *(const float4*)(p + base + c);
        float4 nv = *(const float4*)(n + base + c);
        float d;
        d = av.x - pv.x + 1e-6f; sp += d * d;
        d = av.y - pv.y + 1e-6f; sp += d * d;
        d = av.z - pv.z + 1e-6f; sp += d * d;
        d = av.w - pv.w + 1e-6f; sp += d * d;
        d = av.x - nv.x + 1e-6f; sn += d * d;
        d = av.y - nv.y + 1e-6f; sn += d * d;
        d = av.z - nv.z + 1e-6f; sn += d * d;
        d = av.w - nv.w + 1e-6f; sn += d * d;
    }
    #pragma unroll
    for (int off = 16; off > 0; off >>= 1) {
        sp += __shfl_xor(sp, off, 32);
        sn += __shfl_xor(sn, off, 32);
    }
    if (lane == 0) {
        float l = fmaxf(sqrtf(sp) - sqrtf(sn) + 1.0f, 0.0f);
        atomicAdd(tsum, l);
    }
}

// ---------------------------------------------------------------------------
// Local preservation loss. One block = 16-row strip; 4 waves sweep the 256
// column tiles with V_WMMA_F32_16X16X4_F32 (f32 precision: distances feed a
// top-k ranking, bf16 error ~0.1 absolute would corrupt neighbor selection).
// A-strip staged in LDS; B tiles stream from global (X is 8MB, L2-resident).
// Streaming top-6 + first-11-columns avoids materializing the 64MB matrix.
// ---------------------------------------------------------------------------
__global__ __launch_bounds__(128) void local_loss_kernel(const float* __restrict__ emb,
                                                         const float* __restrict__ sq,
                                                         float* __restrict__ lsum) {
    __shared__ float sA[16 * SA_STRIDE];   // 33.0 KB A strip
    __shared__ float sStage[4][16][16];    // per-wave 16x16 dist tile
    __shared__ float sTopD[4][16][6];
    __shared__ int   sTopI[4][16][6];
    __shared__ float sFirst[16][12];       // dists to columns 0..10

    const int tid  = threadIdx.x;
    const int lane = tid & 31;
    const int wave = tid >> 5;
    const int m0   = blockIdx.x * 16;
    const int nIdx = lane & 15;   // N column within tile (and M row for A frag)
    const int kh   = lane >> 4;   // K-half selector (WMMA f32 16x16x4 layout)

    // Cooperative load of the 16x512 A strip (float4, coalesced).
    for (int i = tid; i < 16 * 128; i += 128) {
        const int row = i >> 7;
        const int c4  = (i & 127) << 2;
        float4 v = *(const float4*)(emb + (size_t)(m0 + row) * DIM + c4);
        *(float4*)(&sA[row * SA_STRIDE + c4]) = v;
    }

    // sq for this strip's rows, per the C-matrix layout: m = r + 8*kh.
    float sqm[8];
    #pragma unroll
    for (int r = 0; r < 8; ++r) sqm[r] = sq[m0 + r + 8 * kh];

    // Register-resident sorted top-6 (ascending) per owned row (lanes 0..15).
    float td[6];
    int   ti[6];
    #pragma unroll
    for (int s = 0; s < 6; ++s) { td[s] = 3.0e38f; ti[s] = -1; }

    __syncthreads();

    const float* ap = &sA[nIdx * SA_STRIDE + 2 * kh];

    for (int t = 0; t < 64; ++t) {
        const int j0 = (t * 4 + wave) * 16;
        const float* bp = emb + (size_t)(j0 + nIdx) * DIM + 2 * kh;

        v8f acc = {};
        #pragma unroll 8
        for (int kb = 0; kb < DIM; kb += 4) {
            v2f a = *(const v2f*)(ap + kb);
            v2f b = *(const v2f*)(bp + kb);
            // D = A(16x4,f32) x B(4x16,f32) + C  -> v_wmma_f32_16x16x4_f32
            acc = __builtin_amdgcn_wmma_f32_16x16x4_f32(
                false, a, false, b, (short)0, acc, false, false);
        }

        const float sqn = sq[j0 + nIdx];
        #pragma unroll
        for (int r = 0; r < 8; ++r) {
            float d2   = sqm[r] + sqn - 2.0f * acc[r];
            float dist = sqrtf(fmaxf(d2, 0.0f) + 1e-12f);
            sStage[wave][r + 8 * kh][nIdx] = dist;
        }
        __syncthreads();

        if (lane < 16) {
            const int row = lane;
            #pragma unroll
            for (int n = 0; n < 16; ++n) {
                float cv = sStage[wave][row][n];
                int   cj = j0 + n;
                #pragma unroll
                for (int s = 0; s < 6; ++s) {   // conditional-swap insert, stays in regs
                    if (cv < td[s]) {
                        float tv = td[s]; int tj = ti[s];
                        td[s] = cv; ti[s] = cj;
                        cv = tv; cj = tj;
                    }
                }
            }
            if (wave == 0 && t == 0) {          // tile j0==0: save columns 0..10
                for (int n = 0; n < 11; ++n) sFirst[row][n] = sStage[0][row][n];
            }
        }
        __syncthreads();
    }

    if (lane < 16) {
        #pragma unroll
        for (int s = 0; s < 6; ++s) {
            sTopD[wave][lane][s] = td[s];
            sTopI[wave][lane][s] = ti[s];
        }
    }
    __syncthreads();

    if (tid < 16) {
        const int row  = tid;
        const int rowG = m0 + row;

        // Merge 4 waves x 6 candidates -> global top-6 (entry 0 = self).
        unsigned used = 0;
        float nd[6]; int ni[6];
        #pragma unroll
        for (int r = 0; r < 6; ++r) {
            float best = 3.4e38f; int bc = 0;
            for (int c = 0; c < 24; ++c) {
                if (used & (1u << c)) continue;
                float v = sTopD[c / 6][row][c % 6];
                if (v < best) { best = v; bc = c; }
            }
            used |= 1u << bc;
            nd[r] = best;
            ni[r] = sTopI[bc / 6][row][bc % 6];
        }

        float nbsum = nd[1] + nd[2] + nd[3] + nd[4] + nd[5];

        // First 5 columns (in index order) that are neither self nor a neighbor:
        // always found within columns 0..10 (exclusion set has <= 6 members).
        float nnsum = 0.0f; int cnt = 0;
        for (int jj = 0; jj < 11 && cnt < 5; ++jj) {
            if (jj == rowG) continue;
            bool isnb = false;
            #pragma unroll
            for (int r = 1; r < 6; ++r) isnb |= (ni[r] == jj);
            if (isnb) continue;
            nnsum += sFirst[row][jj];
            ++cnt;
        }

        float row_loss = fmaxf(nbsum * 0.2f - nnsum * 0.2f + 0.1f, 0.0f);
        atomicAdd(lsum, row_loss);
    }
}

__global__ void final_kernel(const float* __restrict__ ws, float* __restrict__ out) {
    out[0] = ws[0] * (1.0f / B_ROWS) + 0.3f * (ws[1] * (1.0f / B_ROWS));
}

extern "C" void kernel_launch(void* const* d_in, const int* in_sizes, int n_in,
                              void* d_out, int out_size, void* d_ws, size_t ws_size,
                              hipStream_t stream) {
    (void)in_sizes; (void)n_in; (void)out_size; (void)ws_size;
    const float* anchor   = (const float*)d_in[0];
    const float* positive = (const float*)d_in[1];
    const float* negative = (const float*)d_in[2];
    const float* emb      = (const float*)d_in[3];
    float* ws  = (float*)d_ws;          // [0]=triplet sum, [1]=local sum, [2..]=sq[4096]
    float* out = (float*)d_out;

    zero_kernel<<<1, 1, 0, stream>>>(ws);
    sq_kernel<<<B_ROWS / 8, 256, 0, stream>>>(emb, ws + 2);
    triplet_kernel<<<B_ROWS / 8, 256, 0, stream>>>(anchor, positive, negative, ws);
    local_loss_kernel<<<B_ROWS / 16, 128, 0, stream>>>(emb, ws + 2, ws + 1);
    final_kernel<<<1, 1, 0, stream>>>(ws, out);
}